// BPNN_17489106829653
// MI455X (gfx1250) — compile-verified
//
#include <hip/hip_runtime.h>
#include <hip/hip_bf16.h>

typedef float v2f __attribute__((ext_vector_type(2)));
typedef float v4f __attribute__((ext_vector_type(4)));
typedef float v8f __attribute__((ext_vector_type(8)));

// Problem constants (match reference)
constexpr int NB = 512;    // batch
constexpr int NA = 2048;   // atoms
constexpr int F  = 64;
constexpr int H1 = 64;
constexpr int H2 = 32;
constexpr int BT = 128;    // batch rows per block

// LDS layout (dwords). Row stride 68 (= 64 + 4 pad, multiple of 4 for b128/b64
// alignment, and rotates banks by 4 per row -> conflict-free 16-lane column reads).
constexpr int XS   = 0;                 // x tile [BT][68]; reused as h1 after GEMM1
constexpr int H1S  = XS;
constexpr int W0S  = XS + BT * 68;      // 8704: W0 [64][68]
constexpr int W1S  = W0S + H1 * 68;     // 13056: W1 [32][68]
constexpr int REG2_END = W1S + H2 * 68; // 15232
constexpr int H2S  = W0S;               // h2 [BT][36] overlaps W0/W1 (barrier-protected)
constexpr int B0S  = REG2_END;          // 64
constexpr int B1S  = B0S + H1;          // 32
constexpr int W2S  = B1S + H2;          // 32
constexpr int B2S  = W2S + H2;          // 1
constexpr int SMEM_DW = B2S + 4;        // 15364 dwords = 61456 B (fits default 64KB)

__device__ __forceinline__ float silu(float v) {
    return v / (1.0f + __expf(-v));
}

__global__ __launch_bounds__(256, 2)
void bpnn_mlp_kernel(const float* __restrict__ x, const int* __restrict__ types,
                     const float* __restrict__ w0, const float* __restrict__ b0,
                     const float* __restrict__ w1, const float* __restrict__ b1,
                     const float* __restrict__ w2, const float* __restrict__ b2,
                     float* __restrict__ ws) {
    extern __shared__ float sm[];
    const int n     = blockIdx.x;            // atom
    const int brow0 = blockIdx.y * BT;       // batch-tile base
    const int tid   = threadIdx.x;
    const int t     = types[n];              // uniform per block -> scalar

    // ---- Stage x tile [BT x F] (coalesced float4: 16 lanes per row) ----
    #pragma unroll
    for (int it = 0; it < 8; ++it) {
        int q  = it * 256 + tid;             // 2048 float4 slots
        int r  = q >> 4;
        int c4 = (q & 15) << 2;
        v4f v = *(const v4f*)(x + ((size_t)(brow0 + r) * NA + n) * F + c4);
        *(v4f*)(&sm[XS + r * 68 + c4]) = v;
    }
    // ---- Stage W0 [64x64] of this type ----
    const float* w0t = w0 + (size_t)t * H1 * F;
    #pragma unroll
    for (int it = 0; it < 4; ++it) {
        int q = it * 256 + tid;              // 1024 float4 slots
        int r = q >> 4, c4 = (q & 15) << 2;
        *(v4f*)(&sm[W0S + r * 68 + c4]) = *(const v4f*)(w0t + r * F + c4);
    }
    // ---- Stage W1 [32x64] ----
    const float* w1t = w1 + (size_t)t * H2 * H1;
    #pragma unroll
    for (int it = 0; it < 2; ++it) {
        int q = it * 256 + tid;              // 512 float4 slots
        int r = q >> 4, c4 = (q & 15) << 2;
        *(v4f*)(&sm[W1S + r * 68 + c4]) = *(const v4f*)(w1t + r * H1 + c4);
    }
    // ---- Biases / final weights ----
    if (tid < H1)                 sm[B0S + tid]      = b0[t * H1 + tid];
    if (tid >= 64 && tid < 96)    sm[B1S + tid - 64] = b1[t * H2 + (tid - 64)];
    if (tid >= 96 && tid < 128)   sm[W2S + tid - 96] = w2[t * H2 + (tid - 96)];
    if (tid == 128)               sm[B2S]            = b2[t];
    __syncthreads();

    const int wv   = tid >> 5;               // wave id = M-tile (0..7)
    const int lane = tid & 31;
    const int l15  = lane & 15;
    const int khi  = (lane >> 4) ? 2 : 0;    // lanes 16-31 hold K+2,K+3
    const int arow = wv * 16 + l15;          // A-fragment row (own M-tile only)

    // ================= GEMM1: h1 = silu(x * W0^T + b0), [BT x 64] ============
    v8f acc0 = {}, acc1 = {}, acc2 = {}, acc3 = {};
    #pragma unroll
    for (int k = 0; k < 16; ++k) {
        int ko = 4 * k + khi;
        v2f a  = *(const v2f*)(&sm[XS  + arow * 68 + ko]);
        v2f bA = *(const v2f*)(&sm[W0S + ( 0 + l15) * 68 + ko]);   // B[k][nn]=W0[nn][k]
        v2f bB = *(const v2f*)(&sm[W0S + (16 + l15) * 68 + ko]);
        v2f bC = *(const v2f*)(&sm[W0S + (32 + l15) * 68 + ko]);
        v2f bD = *(const v2f*)(&sm[W0S + (48 + l15) * 68 + ko]);
        acc0 = __builtin_amdgcn_wmma_f32_16x16x4_f32(false, a, false, bA, (short)0, acc0, false, false);
        acc1 = __builtin_amdgcn_wmma_f32_16x16x4_f32(false, a, false, bB, (short)0, acc1, false, false);
        acc2 = __builtin_amdgcn_wmma_f32_16x16x4_f32(false, a, false, bC, (short)0, acc2, false, false);
        acc3 = __builtin_amdgcn_wmma_f32_16x16x4_f32(false, a, false, bD, (short)0, acc3, false, false);
    }
    __syncthreads();   // all GEMM1 LDS reads done before h1 overwrites x region

    // C/D layout: VGPR r -> (M = r (+8 for hi lanes), N = l15)
    const int rowoff = wv * 16 + ((lane >> 4) ? 8 : 0);
    #pragma unroll
    for (int r = 0; r < 8; ++r) {
        int row = rowoff + r;
        float v0 = acc0[r] + sm[B0S +  0 + l15];
        float v1 = acc1[r] + sm[B0S + 16 + l15];
        float v2v = acc2[r] + sm[B0S + 32 + l15];
        float v3 = acc3[r] + sm[B0S + 48 + l15];
        sm[H1S + row * 68 +  0 + l15] = silu(v0);
        sm[H1S + row * 68 + 16 + l15] = silu(v1);
        sm[H1S + row * 68 + 32 + l15] = silu(v2v);
        sm[H1S + row * 68 + 48 + l15] = silu(v3);
    }

    // ================= GEMM2: h2 = silu(h1 * W1^T + b1), [BT x 32] ===========
    v8f d0 = {}, d1 = {};
    #pragma unroll
    for (int k = 0; k < 16; ++k) {
        int ko = 4 * k + khi;
        v2f a  = *(const v2f*)(&sm[H1S + arow * 68 + ko]);        // own rows (RAW in-wave)
        v2f bA = *(const v2f*)(&sm[W1S + ( 0 + l15) * 68 + ko]);
        v2f bB = *(const v2f*)(&sm[W1S + (16 + l15) * 68 + ko]);
        d0 = __builtin_amdgcn_wmma_f32_16x16x4_f32(false, a, false, bA, (short)0, d0, false, false);
        d1 = __builtin_amdgcn_wmma_f32_16x16x4_f32(false, a, false, bB, (short)0, d1, false, false);
    }
    __syncthreads();   // all GEMM2 reads of W1 done before h2 overwrites W0/W1 region

    #pragma unroll
    for (int r = 0; r < 8; ++r) {
        int row = rowoff + r;
        float v0 = d0[r] + sm[B1S +  0 + l15];
        float v1 = d1[r] + sm[B1S + 16 + l15];
        sm[H2S + row * 36 +  0 + l15] = silu(v0);
        sm[H2S + row * 36 + 16 + l15] = silu(v1);
    }
    __syncthreads();

    // ================= Final: e = h2 . w2 + b2 -> ws[b*NA + n] ===============
    if (tid < BT) {
        float e = sm[B2S];
        #pragma unroll
        for (int g = 0; g < H2; ++g)
            e += sm[H2S + tid * 36 + g] * sm[W2S + g];
        ws[(size_t)(brow0 + tid) * NA + n] = e;
    }
}

// Deterministic fixed-shape reduction over atoms: out[b] = sum_n ws[b*NA+n]
__global__ __launch_bounds__(256)
void bpnn_reduce_kernel(const float* __restrict__ ws, float* __restrict__ out) {
    __shared__ float red[256];
    const int b = blockIdx.x;
    const int tid = threadIdx.x;
    float s = 0.0f;
    #pragma unroll
    for (int i = 0; i < NA / 256; ++i)
        s += ws[(size_t)b * NA + i * 256 + tid];
    red[tid] = s;
    __syncthreads();
    for (int off = 128; off > 0; off >>= 1) {
        if (tid < off) red[tid] += red[tid + off];
        __syncthreads();
    }
    if (tid == 0) out[b] = red[0];
}

extern "C" void kernel_launch(void* const* d_in, const int* in_sizes, int n_in,
                              void* d_out, int out_size, void* d_ws, size_t ws_size,
                              hipStream_t stream) {
    (void)in_sizes; (void)n_in; (void)out_size; (void)ws_size;
    const float* x     = (const float*)d_in[0];
    const int*   types = (const int*)d_in[1];
    const float* w0    = (const float*)d_in[2];
    const float* b0    = (const float*)d_in[3];
    const float* w1    = (const float*)d_in[4];
    const float* b1    = (const float*)d_in[5];
    const float* w2    = (const float*)d_in[6];
    const float* b2    = (const float*)d_in[7];
    float*       out   = (float*)d_out;
    float*       ws    = (float*)d_ws;   // needs NB*NA*4 = 4 MB

    dim3 grid1(NA, NB / BT);   // 2048 x 4 blocks
    bpnn_mlp_kernel<<<grid1, 256, SMEM_DW * 4, stream>>>(x, types, w0, b0, w1, b1, w2, b2, ws);
    bpnn_reduce_kernel<<<NB, 256, 0, stream>>>(ws, out);
}